// GraphConvWL_16793322127387
// MI455X (gfx1250) — compile-verified
//
#include <hip/hip_runtime.h>
#include <stdint.h>

typedef float v2f __attribute__((ext_vector_type(2)));
typedef float v8f __attribute__((ext_vector_type(8)));

#define DIM 128
#define TILE_ROWS 16
#define GEMM_THREADS 256   // 8 waves

// ---------------- zero workspace ----------------
__global__ void GraphConv_zero_kernel(float* __restrict__ p, size_t n) {
    size_t i = (size_t)blockIdx.x * blockDim.x + threadIdx.x;
    size_t stride = (size_t)gridDim.x * blockDim.x;
    for (; i < n; i += stride) p[i] = 0.0f;
}

// ---------------- edge scatter-add: agg[dst] += feat[src] ----------------
// One wave32 per edge; each lane moves 4 consecutive floats (float4 load +
// 4 global_atomic_add_f32 into L2-resident agg).
__global__ void GraphConv_scatter_kernel(const float* __restrict__ feat,
                                         const long long* __restrict__ src,
                                         const long long* __restrict__ dst,
                                         float* __restrict__ agg,
                                         int nEdges) {
    size_t tid = (size_t)blockIdx.x * blockDim.x + threadIdx.x;
    int e    = (int)(tid >> 5);
    int lane = (int)(tid & 31);
    if (e >= nEdges) return;
    long long s = src[e];
    long long d = dst[e];
    const float4 v = *(const float4*)(feat + (size_t)s * DIM + lane * 4);
    float* ap = agg + (size_t)d * DIM + lane * 4;
    atomicAdd(ap + 0, v.x);
    atomicAdd(ap + 1, v.y);
    atomicAdd(ap + 2, v.z);
    atomicAdd(ap + 3, v.w);
}

// Async-copy one 16x128 f32 tile of agg and feat into LDS (8 KB each).
// 256 threads x 2 chunks x 2 matrices => 4 async b128 ops per thread
// (= ASYNCcnt +4 per wave).
__device__ __forceinline__ void async_copy_tile(const float* __restrict__ agg,
                                                const float* __restrict__ feat,
                                                float* ldsA, float* ldsF,
                                                int m0, int nNodes, int tid) {
#pragma unroll
    for (int k = 0; k < 2; ++k) {
        const int f    = tid + k * GEMM_THREADS;  // float4 index, 512 per tile
        const int row  = f >> 5;                  // 32 float4 per 128-f row
        const int col4 = f & 31;
        int grow = m0 + row;
        if (grow >= nNodes) grow = nNodes - 1;    // clamp (tail safety)
        const float4* gA = (const float4*)(agg  + (size_t)grow * DIM) + col4;
        const float4* gF = (const float4*)(feat + (size_t)grow * DIM) + col4;
        // AS3 pointer value == LDS byte offset (low 32 bits of generic ptr)
        unsigned la = (unsigned)(unsigned long long)(ldsA + f * 4);
        unsigned lf = (unsigned)(unsigned long long)(ldsF + f * 4);
        asm volatile("global_load_async_to_lds_b128 %0, %1, off"
                     :: "v"(la), "v"(gA) : "memory");
        asm volatile("global_load_async_to_lds_b128 %0, %1, off"
                     :: "v"(lf), "v"(gF) : "memory");
    }
}

// ---------------- fused dual GEMM + bias via V_WMMA_F32_16X16X4_F32 ----------
// out = agg @ W_neigh + feat @ W_self + b_neigh
// Persistent blocks: 8 waves each; wave w owns output columns [w*16, w*16+16).
// Each wave keeps its FULL K=128 B-strips of W_neigh and W_self in VGPRs
// (2 x 32 x v2f = 128 VGPRs), loaded once. Row tiles of agg/feat are staged
// through LDS with async global->LDS copies (double-buffered, ASYNCcnt).
__global__ __launch_bounds__(GEMM_THREADS) void GraphConv_gemm_kernel(
    const float* __restrict__ agg,  const float* __restrict__ feat,
    const float* __restrict__ Wn,   const float* __restrict__ Ws,
    const float* __restrict__ bn,   float* __restrict__ out,
    int nNodes, int rowTiles) {
    __shared__ __align__(16) float lds[2][2][TILE_ROWS * DIM];  // 32 KB

    const int tid  = threadIdx.x;
    const int wave = tid >> 5;
    const int lane = tid & 31;
    const int half = lane >> 4;      // K-pair select (A/B), row-half of C/D
    const int ml   = lane & 15;      // M index for A, N index for B/C/D
    const int n0   = wave * 16;      // this wave's 16-column strip

    // ---- preload B fragments for the whole K range (stays in VGPRs) ----
    v2f bWn[32], bWs[32];
#pragma unroll
    for (int i = 0; i < 32; ++i) {
        const int ka = 4 * i + 2 * half;
        bWn[i].x = Wn[(size_t)ka * DIM + n0 + ml];
        bWn[i].y = Wn[(size_t)(ka + 1) * DIM + n0 + ml];
        bWs[i].x = Ws[(size_t)ka * DIM + n0 + ml];
        bWs[i].y = Ws[(size_t)(ka + 1) * DIM + n0 + ml];
    }
    const float bias = bn[n0 + ml];

    // ---- persistent row-tile loop with double-buffered async staging ----
    int t = blockIdx.x;
    if (t < rowTiles) {
        async_copy_tile(agg, feat, &lds[0][0][0], &lds[0][1][0],
                        t * TILE_ROWS, nNodes, tid);
    }
    int nextT = t + gridDim.x;
    int j = 0;
    for (; t < rowTiles; t = nextT, nextT += gridDim.x, ++j) {
        const int cur = j & 1;
        const bool hasNext = nextT < rowTiles;
        if (hasNext) {
            async_copy_tile(agg, feat, &lds[cur ^ 1][0][0], &lds[cur ^ 1][1][0],
                            nextT * TILE_ROWS, nNodes, tid);
            asm volatile("s_wait_asynccnt 0x4" ::: "memory");
        } else {
            asm volatile("s_wait_asynccnt 0x0" ::: "memory");
        }
        __syncthreads();              // tile t visible to all waves

        v8f acc = {};
#pragma unroll
        for (int i = 0; i < 32; ++i) {
            const int ka = 4 * i + 2 * half;
            const v2f aA = *(const v2f*)&lds[cur][0][ml * DIM + ka];
            const v2f aF = *(const v2f*)&lds[cur][1][ml * DIM + ka];
            acc = __builtin_amdgcn_wmma_f32_16x16x4_f32(false, aA, false, bWn[i],
                                                        (short)0, acc, false, false);
            acc = __builtin_amdgcn_wmma_f32_16x16x4_f32(false, aF, false, bWs[i],
                                                        (short)0, acc, false, false);
        }

        const int m0 = t * TILE_ROWS;
#pragma unroll
        for (int r = 0; r < 8; ++r) {
            const int m = m0 + r + 8 * half;
            if (m < nNodes) out[(size_t)m * DIM + n0 + ml] = acc[r] + bias;
        }
        __syncthreads();              // all reads of lds[cur] done before reuse
    }
}

extern "C" void kernel_launch(void* const* d_in, const int* in_sizes, int n_in,
                              void* d_out, int out_size, void* d_ws, size_t ws_size,
                              hipStream_t stream) {
    const float*      feat = (const float*)d_in[0];
    const long long*  src  = (const long long*)d_in[1];   // int64 per reference
    const long long*  dst  = (const long long*)d_in[2];
    const float*      Wn   = (const float*)d_in[3];
    const float*      bn   = (const float*)d_in[4];
    const float*      Ws   = (const float*)d_in[5];
    float*            out  = (float*)d_out;

    const int nNodes = in_sizes[0] / DIM;
    const int nEdges = in_sizes[1];

    float* agg = (float*)d_ws;                 // nNodes*DIM floats (51.2 MB)
    const size_t aggN = (size_t)nNodes * DIM;

    GraphConv_zero_kernel<<<2048, 256, 0, stream>>>(agg, aggN);

    const size_t scatterThreads = (size_t)nEdges * 32;
    const int scatterBlocks = (int)((scatterThreads + 255) / 256);
    GraphConv_scatter_kernel<<<scatterBlocks, 256, 0, stream>>>(feat, src, dst,
                                                                agg, nEdges);

    const int rowTiles = (nNodes + TILE_ROWS - 1) / TILE_ROWS;
    int gemmBlocks = rowTiles < 512 ? rowTiles : 512;
    GraphConv_gemm_kernel<<<gemmBlocks, GEMM_THREADS, 0, stream>>>(
        agg, feat, Wn, Ws, bn, out, nNodes, rowTiles);
}